// Postprocessor_41016937676872
// MI455X (gfx1250) — compile-verified
//
#include <hip/hip_runtime.h>

// SSD post-processor for MI455X (gfx1250, wave32).
// Pipeline: [K1] fused softmax+decode  -> ws
//           [K2] streaming bitonic top-256 per (img,class) in LDS
//           [K3] single-wave NMS: WMMA union-sums + ballot-built 256x256
//                suppression bitmatrix + shuffle-driven greedy scan.

#define B_IMG 32
#define P_NUM 8732
#define C_NUM 21
#define CF    20            // foreground classes
#define KPRE  256
#define TOPK  200
#define CONF_T 0.01f
#define NMS_T  0.45f

typedef float v2f __attribute__((ext_vector_type(2)));
typedef float v8f __attribute__((ext_vector_type(8)));

__device__ __forceinline__ unsigned wave_ballot32(bool pred) {
#if __has_builtin(__builtin_amdgcn_ballot_w32)
    return __builtin_amdgcn_ballot_w32(pred);   // consumes the compare mask
#else
    return (unsigned)__ballot((int)pred);
#endif
}

// ---------------------------------------------------------------- kernel 1
// Fused softmax (max-subtracted, hardware v_exp_f32) + box decode.
// Bandwidth bound: 23.5MB conf in, 22.4MB probs out -> ~2us at 23.3TB/s.
// probs written transposed [B][CF][P] so kernel 2 reads are unit-stride.
__global__ __launch_bounds__(256) void decode_softmax_kernel(
    const float* __restrict__ loc, const float* __restrict__ conf,
    const float* __restrict__ prior, float* __restrict__ boxes,
    float* __restrict__ probs)
{
    const int p = blockIdx.x * 256 + threadIdx.x;
    const int b = blockIdx.y;
    if (p >= P_NUM) return;

    const float* cr = conf + ((size_t)b * P_NUM + p) * C_NUM;
    float v[C_NUM];
    #pragma unroll
    for (int c = 0; c < C_NUM; ++c) v[c] = cr[c];
    float mx = v[0];
    #pragma unroll
    for (int c = 1; c < C_NUM; ++c) mx = fmaxf(mx, v[c]);
    float s = 0.f;
    #pragma unroll
    for (int c = 0; c < C_NUM; ++c) { v[c] = __expf(v[c] - mx); s += v[c]; }
    const float inv = 1.0f / s;
    #pragma unroll
    for (int c = 1; c < C_NUM; ++c)
        probs[((size_t)b * CF + (c - 1)) * P_NUM + p] = v[c] * inv;

    const float4 l4 = reinterpret_cast<const float4*>(loc)[(size_t)b * P_NUM + p];
    const float4 pr = reinterpret_cast<const float4*>(prior)[p];
    const float cx = pr.x + l4.x * 0.1f * pr.z;
    const float cy = pr.y + l4.y * 0.1f * pr.w;
    const float w  = pr.z * __expf(l4.z * 0.2f);
    const float h  = pr.w * __expf(l4.w * 0.2f);
    float4 bx;
    bx.x = cx - 0.5f * w;  bx.y = cy - 0.5f * h;
    bx.z = bx.x + w;       bx.w = bx.y + h;
    reinterpret_cast<float4*>(boxes)[(size_t)b * P_NUM + p] = bx;
}

// ---------------------------------------------------------------- kernel 2
// Streaming bitonic top-256 of 8732 per (img,class). 4KB LDS: a 512-entry
// buffer holding (score_bits<<32 | ~idx) keys. Each round: sort new 256
// ascending, bitonic-merge against the running descending top-256.
__global__ __launch_bounds__(256) void topk_kernel(
    const float* __restrict__ probs, const float* __restrict__ boxes,
    float* __restrict__ cscore, float* __restrict__ cbox)
{
    __shared__ unsigned long long buf[512];
    const int bc  = blockIdx.x;          // b*CF + c
    const int b   = bc / CF;
    const int tid = threadIdx.x;
    const float* src = probs + (size_t)bc * P_NUM;

    auto loadkey = [&](int p) -> unsigned long long {
        if (p < P_NUM) {
            const float sv = src[p];     // softmax prob > 0 -> bits monotone
            return ((unsigned long long)__float_as_uint(sv) << 32) |
                   (unsigned long long)(0xFFFFFFFFu - (unsigned)p); // tie: low idx first
        }
        return 0ull;
    };
    auto cswap = [&](int i, int q, bool desc) {
        unsigned long long a = buf[i], c = buf[q];
        const bool sw = desc ? (a < c) : (a > c);
        if (sw) { buf[i] = c; buf[q] = a; }
    };
    auto sort256 = [&](int base, bool dir) {   // dir=true -> descending
        for (int k = 2; k <= 256; k <<= 1)
            for (int j = k >> 1; j > 0; j >>= 1) {
                __syncthreads();
                if (tid < 128) {
                    const int i = ((tid & ~(j - 1)) << 1) | (tid & (j - 1));
                    const int q = i | j;
                    const bool desc = ((i & k) == 0) ? dir : !dir;
                    cswap(base + i, base + q, desc);
                }
            }
        __syncthreads();
    };

    buf[tid] = loadkey(tid);
    sort256(0, true);
    const int nch = (P_NUM + 255) / 256;       // 35
    for (int r = 1; r < nch; ++r) {
        const int p = r * 256 + tid;
        if (p + 256 < P_NUM) __builtin_prefetch(src + p + 256, 0, 1); // global_prefetch_b8
        buf[256 + tid] = loadkey(p);
        sort256(256, false);                   // back half ascending
        for (int j = 256; j > 0; j >>= 1) {    // merge desc||asc -> desc
            __syncthreads();
            const int i = ((tid & ~(j - 1)) << 1) | (tid & (j - 1));
            cswap(i, i | j, true);
        }
        __syncthreads();
    }

    const unsigned long long key = buf[tid];
    const float sv = __uint_as_float((unsigned)(key >> 32));
    unsigned idx = 0xFFFFFFFFu - (unsigned)(key & 0xFFFFFFFFull);
    if (idx >= P_NUM) idx = P_NUM - 1;         // padding keys (score 0)
    cscore[(size_t)bc * KPRE + tid] = sv;
    reinterpret_cast<float4*>(cbox)[(size_t)bc * KPRE + tid] =
        reinterpret_cast<const float4*>(boxes)[(size_t)b * P_NUM + idx];
}

// ---------------------------------------------------------------- kernel 3
// One wave32 per (img,class). area_i+area_j per 16x16 tile on the matrix
// pipe (v_wmma_f32_16x16x4_f32, two K-uniform rank-1 products -> layout-
// robust; XDL co-executes with the VALU min/max intersection math).
// IoU test is division-free: inter/den > T  <=>  inter > T*den (den>0).
// The j>i causality constraint is free on off-diagonal tiles (ti<tj ⇒
// always true) and a compile-time SALU mask on the diagonal tile.
// Suppression bytes are packed into a 64-bit register across the unrolled
// row loop and stored with ONE ds_store_b64 per tile by lanes 0/8/16/24
// (whose required mask shift conveniently equals the lane id).
__global__ __launch_bounds__(32) void nms_kernel(
    const float* __restrict__ cscore, const float* __restrict__ cbox,
    float* __restrict__ out)
{
    __shared__ float s_score[KPRE];
    __shared__ float s_box[KPRE * 4];
    __shared__ float s_area[KPRE];
    __shared__ uint4 s_sup4[512];              // 8KB: supT[lane][row] bytes
    const int bc   = blockIdx.x;
    const int lane = threadIdx.x;

    for (int c = lane; c < KPRE; c += 32) {
        const float sv = cscore[(size_t)bc * KPRE + c];
        const float4 bx = reinterpret_cast<const float4*>(cbox)[(size_t)bc * KPRE + c];
        s_score[c] = sv;
        s_box[c * 4 + 0] = bx.x; s_box[c * 4 + 1] = bx.y;
        s_box[c * 4 + 2] = bx.z; s_box[c * 4 + 3] = bx.w;
        s_area[c] = fmaxf(bx.z - bx.x, 0.f) * fmaxf(bx.w - bx.y, 0.f);
    }
    for (int t = lane; t < 512; t += 32) { uint4 z; z.x=z.y=z.z=z.w=0u; s_sup4[t] = z; }
    __syncthreads();

    unsigned char* supB = reinterpret_cast<unsigned char*>(s_sup4);
    const int m    = lane & 15;                // N index / A-row index
    const int half = lane >> 4;
    // Writer decomposition: lanes with (lane&7)==0 own byte s=(lane>>3)&1 of
    // row-half h=lane>>4; its bit position in the ballot mask is 16h+8s==lane.
    const bool writer = ((lane & 7) == 0);
    const int  wsub   = ((lane >> 3) & 1);     // which byte of the 16-bit seg
    const int  wrowof = (half << 3);           // row offset within tile

    auto tile = [&](int ti, int tj, bool diag) {
        const int j = tj * 16 + m;
        v2f a1; a1.x = s_area[ti * 16 + m]; a1.y = a1.x;    // A row m = area_i (all K)
        const float ajv = s_area[j];
        v2f b2; b2.x = ajv; b2.y = ajv;                     // B col n = area_j (all K)
        v2f q;  q.x = 0.25f; q.y = 0.25f;                   // 4 K-slots * 0.25 = 1
        v8f acc = {};
        acc = __builtin_amdgcn_wmma_f32_16x16x4_f32(false, a1, false, q,
                                                    (short)0, acc, false, false);
        acc = __builtin_amdgcn_wmma_f32_16x16x4_f32(false, q, false, b2,
                                                    (short)0, acc, false, false);
        // acc[r] (lane) = area_{ti*16+r+8*half} + area_{tj*16+m}
        const float bj0 = s_box[j*4+0], bj1 = s_box[j*4+1];
        const float bj2 = s_box[j*4+2], bj3 = s_box[j*4+3];
        unsigned lo = 0, hi = 0;                // my 8 suppression bytes
        #pragma unroll
        for (int r = 0; r < 8; ++r) {
            const int i = ti * 16 + r + (half << 3);
            const float bi0 = s_box[i*4+0], bi1 = s_box[i*4+1];
            const float bi2 = s_box[i*4+2], bi3 = s_box[i*4+3];
            const float xx1 = fmaxf(bi0, bj0), yy1 = fmaxf(bi1, bj1);
            const float xx2 = fminf(bi2, bj2), yy2 = fminf(bi3, bj3);
            const float iw = fmaxf(xx2 - xx1, 0.f), ih = fmaxf(yy2 - yy1, 0.f);
            const float inter = iw * ih;
            const float den = fmaxf(acc[r] - inter, 1e-9f);   // union > 0
            unsigned msk = wave_ballot32(inter > NMS_T * den); // wave-uniform
            if (diag) {   // j>i mask, compile-time constant per r (SALU and)
                const unsigned dm = ((0xFFFFu << (r + 1)) & 0xFFFFu) |
                                    (((0xFFFFu << (r + 9)) & 0xFFFFu) << 16);
                msk &= dm;
            }
            const unsigned byte = (msk >> lane) & 0xFFu;      // shift == 16h+8s
            if (r < 4) lo |= byte << (8 * r);
            else       hi |= byte << (8 * (r - 4));
        }
        if (writer) {
            const int off = (2 * tj + wsub) * 256 + ti * 16 + wrowof; // 8B aligned
            *reinterpret_cast<unsigned long long*>(supB + off) =
                ((unsigned long long)hi << 32) | (unsigned long long)lo;
        }
    };

    for (int ti = 0; ti < 16; ++ti) {
        tile(ti, ti, true);                    // diagonal: apply j>i row masks
        for (int tj = ti + 1; tj < 16; ++tj)
            tile(ti, tj, false);               // off-diagonal: j>i always true
    }
    __syncthreads();

    // Greedy scan: lane l owns columns 8l..8l+7; keep bit per column.
    unsigned keep = 0;
    #pragma unroll
    for (int bit = 0; bit < 8; ++bit)
        if (s_score[lane * 8 + bit] > CONF_T) keep |= (1u << bit);

    for (int blk = 0; blk < 16; ++blk) {
        const uint4 w = s_sup4[lane * 16 + blk];   // 16 rows of my byte column
        unsigned words[4] = { w.x, w.y, w.z, w.w };
        #pragma unroll
        for (int t = 0; t < 16; ++t) {
            const int i = blk * 16 + t;
            const unsigned kv = (unsigned)__shfl((int)keep, i >> 3, 32);
            const unsigned ki = (kv >> (i & 7)) & 1u;
            const unsigned row = (words[t >> 2] >> ((t & 3) * 8)) & 0xFFu;
            keep &= ~(row & (0u - ki));            // suppress j>i if keep[i]
        }
    }

    // Compact survivors (already score-ordered) into out[0..199], zero rest.
    const unsigned cnt = __popc(keep);
    unsigned pre = cnt;
    for (int off = 1; off < 32; off <<= 1) {
        const unsigned n = (unsigned)__shfl_up((int)pre, off, 32);
        if (lane >= off) pre += n;
    }
    pre -= cnt;                                    // exclusive prefix

    float* op = out + (size_t)bc * (TOPK * 5);
    for (int t = lane; t < TOPK * 5; t += 32) op[t] = 0.0f;
    asm volatile("s_wait_storecnt 0x0" ::: "memory");  // zero-fill before scatter
    #pragma unroll
    for (int bit = 0; bit < 8; ++bit) {
        if ((keep >> bit) & 1u) {
            const unsigned pos = pre + (unsigned)__popc(keep & ((1u << bit) - 1u));
            if (pos < TOPK) {
                const int c = lane * 8 + bit;
                op[pos*5+0] = s_score[c];
                op[pos*5+1] = s_box[c*4+0];
                op[pos*5+2] = s_box[c*4+1];
                op[pos*5+3] = s_box[c*4+2];
                op[pos*5+4] = s_box[c*4+3];
            }
        }
    }
}

extern "C" void kernel_launch(void* const* d_in, const int* in_sizes, int n_in,
                              void* d_out, int out_size, void* d_ws, size_t ws_size,
                              hipStream_t stream) {
    (void)in_sizes; (void)n_in; (void)out_size; (void)ws_size;
    const float* loc   = (const float*)d_in[0];
    const float* conf  = (const float*)d_in[1];
    const float* prior = (const float*)d_in[2];

    float* ws     = (float*)d_ws;
    float* boxes  = ws;                                  // B*P*4
    float* probs  = boxes  + (size_t)B_IMG * P_NUM * 4;  // B*CF*P
    float* cscore = probs  + (size_t)B_IMG * CF * P_NUM; // B*CF*256
    float* cbox   = cscore + (size_t)B_IMG * CF * KPRE;  // B*CF*256*4

    dim3 g1((P_NUM + 255) / 256, B_IMG);
    decode_softmax_kernel<<<g1, 256, 0, stream>>>(loc, conf, prior, boxes, probs);
    topk_kernel<<<B_IMG * CF, 256, 0, stream>>>(probs, boxes, cscore, cbox);
    nms_kernel<<<B_IMG * CF, 32, 0, stream>>>(cscore, cbox, (float*)d_out);
}